// DNAOrigamiSE3Transformer_14516989460878
// MI455X (gfx1250) — compile-verified
//
#include <hip/hip_runtime.h>
#include <hip/hip_bf16.h>

#define N_NODES 65536
#define E_TOT   1048576
#define HID     80

typedef float v2f __attribute__((ext_vector_type(2)));
typedef float v8f __attribute__((ext_vector_type(8)));

// D = A(16x4 f32) * B(4x16 f32) + C(16x16 f32), wave32
__device__ __forceinline__ v8f wmma4(v2f a, v2f b, v8f c) {
    return __builtin_amdgcn_wmma_f32_16x16x4_f32(
        /*neg_a=*/false, a, /*neg_b=*/false, b,
        /*c_mod=*/(short)0, c, /*reuse_a=*/false, /*reuse_b=*/false);
}

// Internal hidden-state layout is channel-permuted (SoA over vector comps):
//   permuted [0,32)            = scalar channels (same as reference)
//   permuted [32+16k, 48+16k)  = vector component k, multiplicities a=0..15
// Reference channel of permuted index pc:
__device__ __forceinline__ int perm_to_orig(int pc) {
    if (pc < 32) return pc;
    int r = pc - 32;
    int k = r >> 4;
    int a = r & 15;
    return 32 + 3 * a + k;
}

// ---------------------------------------------------------------------------
// h0 = x @ W_in + b_in   (N x 16) @ (16 x 80), written in permuted layout
// ---------------------------------------------------------------------------
__global__ void __launch_bounds__(256) input_kernel(
    const float* __restrict__ x, const float* __restrict__ Wi,
    const float* __restrict__ bi, float* __restrict__ h)
{
    int t = blockIdx.x * blockDim.x + threadIdx.x;   // t < N*80
    int node = t / HID;
    int pc   = t - node * HID;
    int c    = perm_to_orig(pc);
    float acc = bi[c];
    const float* xr = x + node * 16;
#pragma unroll
    for (int i = 0; i < 16; ++i) acc += xr[i] * Wi[i * HID + c];
    h[t] = acc;
}

__global__ void __launch_bounds__(256) zero_kernel(float* __restrict__ p)
{
    p[blockIdx.x * blockDim.x + threadIdx.x] = 0.0f;
}

// ---------------------------------------------------------------------------
// out = relu(h) @ W_out + b_out   (N x 80, permuted) @ (80 x 8)
// ---------------------------------------------------------------------------
__global__ void __launch_bounds__(256) output_kernel(
    const float* __restrict__ h, const float* __restrict__ Wo,
    const float* __restrict__ bo, float* __restrict__ out)
{
    int t = blockIdx.x * blockDim.x + threadIdx.x;   // t < N*8
    int node = t >> 3;
    int c    = t & 7;
    float acc = bo[c];
    const float* hr = h + node * HID;
#pragma unroll
    for (int pc = 0; pc < HID; ++pc) {               // perm_to_orig is constexpr here
        float v = hr[pc];
        v = v > 0.0f ? v : 0.0f;
        acc += v * Wo[perm_to_orig(pc) * 8 + c];
    }
    out[t] = acc;
}

// ---------------------------------------------------------------------------
// Edge kernel: one wave32 processes tiles of 16 edges via f32 WMMA.
//   out_s(16x32) = (s2*S1) @ (cS*w1) + D @ (cD*w2)
//   T3  (16x16)  = S1 @ (cS*w3)
//   T4k (16x16)  = V1k @ (cV*w4),  k=0..2
//   message v-part[k][c] = T3[c]*v2[k] + s2*T4k[c]   (permuted layout)
// All path norms and 1/sqrt(DEG) folded into B fragments.
// ---------------------------------------------------------------------------
__global__ void __launch_bounds__(256) edge_kernel(
    const float* __restrict__ hin, float* __restrict__ hout,
    const int*   __restrict__ eidx, const float* __restrict__ eattr,
    const float* __restrict__ w1, const float* __restrict__ w2,
    const float* __restrict__ w3, const float* __restrict__ w4)
{
    const int lane  = threadIdx.x & 31;
    const int hh    = lane >> 4;       // lane half: selects K pair / row group
    const int n     = lane & 15;       // B/C column (edge row for A loads)
    const int waveId = blockIdx.x * (blockDim.x >> 5) + (threadIdx.x >> 5);
    const int nWaves = gridDim.x * (blockDim.x >> 5);
    const int numTiles = E_TOT / 16;

    // fold (1/sqrt(2)) * path-norm * (1/sqrt(DEG)=0.25) into weights
    const float cS = 0.03125f;        // 1/sqrt(2)/sqrt(32)/4 = 1/32
    const float cD = 0.0255155180f;   // 1/sqrt(2)/sqrt(48)/4
    const float cV = 0.0441941738f;   // 1/sqrt(2)/sqrt(16)/4

    // ---- B (weight) fragments, resident in VGPRs for the whole kernel ----
    v2f B1[8][2], B3[8], B2[4][2], B4[4];
#pragma unroll
    for (int c = 0; c < 8; ++c) {
        const int k0 = 4 * c + 2 * hh;
#pragma unroll
        for (int t = 0; t < 2; ++t) {
            v2f b; b.x = w1[k0 * 32 + 16 * t + n] * cS;
                   b.y = w1[(k0 + 1) * 32 + 16 * t + n] * cS;
            B1[c][t] = b;
        }
        v2f b3; b3.x = w3[k0 * 16 + n] * cS;
                b3.y = w3[(k0 + 1) * 16 + n] * cS;
        B3[c] = b3;
    }
#pragma unroll
    for (int c = 0; c < 4; ++c) {
        const int k0 = 4 * c + 2 * hh;
#pragma unroll
        for (int t = 0; t < 2; ++t) {
            v2f b; b.x = w2[k0 * 32 + 16 * t + n] * cD;
                   b.y = w2[(k0 + 1) * 32 + 16 * t + n] * cD;
            B2[c][t] = b;
        }
        v2f b4; b4.x = w4[k0 * 16 + n] * cV;
                b4.y = w4[(k0 + 1) * 16 + n] * cV;
        B4[c] = b4;
    }

    const v8f Z = {0.f, 0.f, 0.f, 0.f, 0.f, 0.f, 0.f, 0.f};

    for (int tile = waveId; tile < numTiles; tile += nWaves) {
        const int eBase = tile * 16;

        // prefetch next tile's edge records into cache
        if (tile + nWaves < numTiles) {
            const int ne = (tile + nWaves) * 16 + n;
            __builtin_prefetch(&eidx[ne], 0, 1);
            __builtin_prefetch(&eattr[4 * ne], 0, 1);
        }

        // this lane's A-row is edge (eBase + n); both lane halves load it
        const int e   = eBase + n;
        const int src = eidx[e];
        const float4 at = *reinterpret_cast<const float4*>(eattr + 4 * e);
        const float s2 = at.x, v2x = at.y, v2y = at.z, v2z = at.w;
        const float* row = hin + src * HID;

        // ---- vector-part A fragments: contiguous pairs in permuted layout
        v2f VA[3][4];
#pragma unroll
        for (int k = 0; k < 3; ++k)
#pragma unroll
            for (int c = 0; c < 4; ++c)
                VA[k][c] = *reinterpret_cast<const v2f*>(row + 32 + 16 * k + 4 * c + 2 * hh);

        // ---- T4k = V1k @ w4 (A fragments used directly, no repacking)
        v8f T40 = Z, T41 = Z, T42 = Z;
#pragma unroll
        for (int c = 0; c < 4; ++c) {
            T40 = wmma4(VA[0][c], B4[c], T40);
            T41 = wmma4(VA[1][c], B4[c], T41);
            T42 = wmma4(VA[2][c], B4[c], T42);
        }

        // ---- d[a] = v1[a]·v2 fragments (packed fma), into out_s via w2
        v8f Cs0 = Z, Cs1 = Z;
#pragma unroll
        for (int c = 0; c < 4; ++c) {
            v2f dd = VA[0][c] * v2x + VA[1][c] * v2y + VA[2][c] * v2z;
            Cs0 = wmma4(dd, B2[c][0], Cs0);
            Cs1 = wmma4(dd, B2[c][1], Cs1);
        }

        // ---- scalar-part: S1 into out_s (scaled by s2) and into T3
        v8f T3 = Z;
#pragma unroll
        for (int c = 0; c < 8; ++c) {
            v2f S = *reinterpret_cast<const v2f*>(row + 4 * c + 2 * hh);  // 8B aligned
            v2f A1 = S * s2;
            Cs0 = wmma4(A1, B1[c][0], Cs0);
            Cs1 = wmma4(A1, B1[c][1], Cs1);
            T3  = wmma4(S, B3[c], T3);
        }

        // ---- scatter: lane holds column n, rows m = 8*hh + v
#pragma unroll
        for (int v = 0; v < 8; ++v) {
            const int em  = eBase + 8 * hh + v;
            const int dst = eidx[E_TOT + em];
            const float4 am = *reinterpret_cast<const float4*>(eattr + 4 * em);
            float* ob = hout + dst * HID;
            atomicAdd(ob + n,      Cs0[v]);
            atomicAdd(ob + 16 + n, Cs1[v]);
            const float t3 = T3[v];
            atomicAdd(ob + 32 + n,      t3 * am.y + am.x * T40[v]);
            atomicAdd(ob + 48 + n,      t3 * am.z + am.x * T41[v]);
            atomicAdd(ob + 64 + n,      t3 * am.w + am.x * T42[v]);
        }
    }
}

// ---------------------------------------------------------------------------
extern "C" void kernel_launch(void* const* d_in, const int* in_sizes, int n_in,
                              void* d_out, int out_size, void* d_ws, size_t ws_size,
                              hipStream_t stream)
{
    const float* x     = (const float*)d_in[0];
    const int*   eidx  = (const int*)  d_in[1];
    const float* eattr = (const float*)d_in[2];
    const float* W_in  = (const float*)d_in[3];
    const float* b_in  = (const float*)d_in[4];
    const float* w1    = (const float*)d_in[5];   // (3,32,32)
    const float* w2    = (const float*)d_in[6];   // (3,16,32)
    const float* w3    = (const float*)d_in[7];   // (3,32,16)
    const float* w4    = (const float*)d_in[8];   // (3,16,16)
    const float* W_out = (const float*)d_in[9];
    const float* b_out = (const float*)d_in[10];
    float* out = (float*)d_out;

    float* hA = (float*)d_ws;
    float* hB = hA + (size_t)N_NODES * HID;

    const int nh = N_NODES * HID;                 // 5,242,880
    input_kernel<<<nh / 256, 256, 0, stream>>>(x, W_in, b_in, hA);

    float* hin = hA;
    float* hout = hB;
    for (int l = 0; l < 3; ++l) {
        zero_kernel<<<nh / 256, 256, 0, stream>>>(hout);
        edge_kernel<<<2048, 256, 0, stream>>>(
            hin, hout, eidx, eattr,
            w1 + l * 32 * 32, w2 + l * 16 * 32,
            w3 + l * 32 * 16, w4 + l * 16 * 16);
        float* t = hin; hin = hout; hout = t;
    }

    output_kernel<<<(N_NODES * 8) / 256, 256, 0, stream>>>(hin, W_out, b_out, out);
}